// ProImg_24292335026253
// MI455X (gfx1250) — compile-verified
//
#include <hip/hip_runtime.h>
#include <hip/hip_bf16.h>

// ---------------------------------------------------------------------------
// ProImg pipeline on MI455X (gfx1250, wave32).
// Memory-bound on the 268MB img read (~11.5us at 23.3 TB/s); all GEMMs fit in
// the 192MB L2 and are compute-trivial (~2.5 GFLOP), so we keep full f32
// precision through the matrix cores via V_WMMA_F32_16X16X4_F32 to preserve
// the lp>ln comparison semantics in the epilogue.
// ---------------------------------------------------------------------------

typedef float v2f __attribute__((ext_vector_type(2)));
typedef float v8f __attribute__((ext_vector_type(8)));

#define BATCH   1024
#define C_IN    2048
#define POOLSZ  32      // 2*4*4 spatial elements per (b, c)

// ---------------------------------------------------------------------------
// Kernel 1: mean over the 32 contiguous spatial elements of each (b, c).
// img is [B, 2048, 2, 4, 4] row-major -> 32 contiguous floats per (b, c).
// One thread per (b, c); 8 x float4 loads -> global_load_b128, streams 268MB.
// ---------------------------------------------------------------------------
__global__ void pool_mean32(const float* __restrict__ img,
                            float* __restrict__ feat, int total) {
    int i = blockIdx.x * blockDim.x + threadIdx.x;
    if (i >= total) return;
    const float4* p = reinterpret_cast<const float4*>(img + (size_t)i * POOLSZ);
    float s = 0.0f;
#pragma unroll
    for (int j = 0; j < 8; ++j) {
        float4 q = p[j];
        s += (q.x + q.y) + (q.z + q.w);
    }
    feat[i] = s * (1.0f / 32.0f);
}

// ---------------------------------------------------------------------------
// Kernel 2: f32 WMMA GEMM, C = act(A[M,K] @ W[K,N] + bias[N]).
// One wave (32 lanes) per 16x16 output tile; K consumed 4 per WMMA.
//
// VGPR layouts per CDNA5 ISA 7.12.2 (32-bit):
//   A 16x4 : lanes 0-15 -> M=lane, K-pair {0,1}; lanes 16-31 -> M=lane-16,
//            K-pair {2,3}. VGPR0 = first K of pair, VGPR1 = second.
//   B 4x16 : symmetric (N = lane&15, K-pair selected by lane[4]).
//   C 16x16: 8 VGPRs; lane gives N = lane&15, rows M = r + 8*(lane>>4).
// ---------------------------------------------------------------------------
template <bool RELU>
__global__ void wmma_gemm_f32(const float* __restrict__ A,
                              const float* __restrict__ W,
                              const float* __restrict__ bias,
                              float* __restrict__ C,
                              int M, int N, int K) {
    const int lane  = threadIdx.x;      // 0..31
    const int half  = lane >> 4;        // 0 or 1 (selects K-pair of fragment)
    const int l15   = lane & 15;
    const int ntile = blockIdx.x;
    const int mtile = blockIdx.y;

    const int n  = ntile * 16 + l15;    // output column owned by this lane
    const int am = mtile * 16 + l15;    // A row this lane fetches

    // Accumulator initialized with bias (same column n for all 8 rows).
    const float bv = bias[n];
    v8f acc;
#pragma unroll
    for (int r = 0; r < 8; ++r) acc[r] = bv;

    const float* __restrict__ Arow = A + (size_t)am * K + 2 * half;   // + k
    const float* __restrict__ Wcol = W + (size_t)(2 * half) * N + n;  // + k*N

    // K is always a multiple of 32 here (2048/512/256/128).
    for (int k = 0; k < K; k += 32) {
#pragma unroll
        for (int kk = 0; kk < 32; kk += 4) {
            v2f a, b;
            a.x = Arow[k + kk];
            a.y = Arow[k + kk + 1];
            b.x = Wcol[(size_t)(k + kk) * N];
            b.y = Wcol[(size_t)(k + kk + 1) * N];
            // (neg_a, A, neg_b, B, c_mod, C, reuse_a, reuse_b)
            acc = __builtin_amdgcn_wmma_f32_16x16x4_f32(
                false, a, false, b, (short)0, acc, false, false);
        }
    }

    float* __restrict__ Cout = C + (size_t)(mtile * 16 + 8 * half) * N + n;
#pragma unroll
    for (int r = 0; r < 8; ++r) {
        float v = acc[r];
        if (RELU) v = fmaxf(v, 0.0f);
        Cout[(size_t)r * N] = v;
    }
}

// ---------------------------------------------------------------------------
// Kernel 3: epilogue. One 128-thread block per batch row.
//   nsp = i_sp/max(||i_sp||,eps); lp = nsp . p+/max(||p+||,eps); ln likewise.
//   is_bad = (lp>ln)==label && lp!=ln; scale = is_bad ? tanh(-|lp-ln|)+1 : 1
//   out[b,0:128] = i_sp*scale ; out[b,128:256] = i_sh
// ---------------------------------------------------------------------------
__device__ __forceinline__ float block_reduce_128(float x, float* red) {
    const int t = threadIdx.x;
    red[t] = x;
    __syncthreads();
#pragma unroll
    for (int s = 64; s > 0; s >>= 1) {
        if (t < s) red[t] += red[t + s];
        __syncthreads();
    }
    float r = red[0];
    __syncthreads();
    return r;
}

__global__ void epilogue(const float* __restrict__ isp,
                         const float* __restrict__ ish,
                         const int* __restrict__ label,
                         const float* __restrict__ ppos,
                         const float* __restrict__ pneg,
                         float* __restrict__ out) {
    __shared__ float red[128];
    const int b = blockIdx.x;
    const int t = threadIdx.x;

    const float v  = isp[b * 128 + t];
    const float pp = ppos[t];
    const float nn = pneg[t];

    const float nv  = block_reduce_128(v * v, red);
    const float npp = block_reduce_128(pp * pp, red);
    const float nnn = block_reduce_128(nn * nn, red);
    const float dp  = block_reduce_128(v * pp, red);
    const float dn  = block_reduce_128(v * nn, red);

    const float eps = 1e-12f;
    const float dv  = fmaxf(sqrtf(nv), eps);
    const float lp  = dp / (dv * fmaxf(sqrtf(npp), eps));
    const float ln  = dn / (dv * fmaxf(sqrtf(nnn), eps));

    const bool lab    = label[b] != 0;
    const bool pred   = lp > ln;
    const bool is_bad = (pred == lab) && (lp != ln);
    const float diff  = fabsf(lp - ln);
    const float scale = is_bad ? (tanhf(-diff) + 1.0f) : 1.0f;

    out[b * 256 + t]       = v * scale;
    out[b * 256 + 128 + t] = ish[b * 128 + t];
}

// ---------------------------------------------------------------------------
// Host-side launcher. Inputs (setup_inputs order):
//  0 img [B,2048,2,4,4] f32   1 label [B] i32
//  2 W1 [2048,512]  3 b1[512] 4 W2[512,256] 5 b2[256]
//  6 W3 [256,128]   7 b3[128] 8 Wsp[128,128] 9 bsp[128]
// 10 Wsh[128,128]  11 bsh[128] 12 pro_pos[128] 13 pro_neg[128]
// Output: features [B, 256] f32.
// ---------------------------------------------------------------------------
extern "C" void kernel_launch(void* const* d_in, const int* in_sizes, int n_in,
                              void* d_out, int out_size, void* d_ws, size_t ws_size,
                              hipStream_t stream) {
    const float* img   = (const float*)d_in[0];
    const int*   label = (const int*)d_in[1];
    const float* W1  = (const float*)d_in[2];
    const float* b1  = (const float*)d_in[3];
    const float* W2  = (const float*)d_in[4];
    const float* b2  = (const float*)d_in[5];
    const float* W3  = (const float*)d_in[6];
    const float* b3  = (const float*)d_in[7];
    const float* Wsp = (const float*)d_in[8];
    const float* bsp = (const float*)d_in[9];
    const float* Wsh = (const float*)d_in[10];
    const float* bsh = (const float*)d_in[11];
    const float* ppn = (const float*)d_in[12];
    const float* pnn = (const float*)d_in[13];

    // Workspace layout (floats): ~12.5 MB total.
    float* ws   = (float*)d_ws;
    float* feat = ws;                              // [1024, 2048]
    float* h1   = feat + (size_t)BATCH * C_IN;     // [1024, 512]
    float* h2   = h1   + (size_t)BATCH * 512;      // [1024, 256]
    float* x    = h2   + (size_t)BATCH * 256;      // [1024, 128]
    float* isp  = x    + (size_t)BATCH * 128;      // [1024, 128]
    float* ish  = isp  + (size_t)BATCH * 128;      // [1024, 128]

    // 1) pooling: streams the full 268MB img tensor once.
    {
        const int total = BATCH * C_IN;
        pool_mean32<<<(total + 255) / 256, 256, 0, stream>>>(img, feat, total);
    }

    // 2) GEMM chain (one wave per 16x16 tile).
    wmma_gemm_f32<true><<<dim3(512 / 16, BATCH / 16), 32, 0, stream>>>(
        feat, W1, b1, h1, BATCH, 512, C_IN);
    wmma_gemm_f32<true><<<dim3(256 / 16, BATCH / 16), 32, 0, stream>>>(
        h1, W2, b2, h2, BATCH, 256, 512);
    wmma_gemm_f32<false><<<dim3(128 / 16, BATCH / 16), 32, 0, stream>>>(
        h2, W3, b3, x, BATCH, 128, 256);

    // 3) two 128->128 heads.
    wmma_gemm_f32<false><<<dim3(128 / 16, BATCH / 16), 32, 0, stream>>>(
        x, Wsp, bsp, isp, BATCH, 128, 128);
    wmma_gemm_f32<false><<<dim3(128 / 16, BATCH / 16), 32, 0, stream>>>(
        x, Wsh, bsh, ish, BATCH, 128, 128);

    // 4) cosine logits + data-dependent scale + concat.
    epilogue<<<BATCH, 128, 0, stream>>>(isp, ish, label, ppn, pnn, (float*)d_out);
}